// Canny_59717225283780
// MI455X (gfx1250) — compile-verified
//
#include <hip/hip_runtime.h>
#include <stdint.h>

#define IMH 1024
#define IMW 1024
#define TS  32
// fused kernel tile sizes (with halos)
#define IN_T 38   // rows staged: TS + 2*3
#define IN_W 40   // floats staged per row, origin gx = tx0-4 (16B aligned chunks)
#define IN_S 44   // LDS row stride (floats); 44*4=176B, multiple of 16
#define SM_T 36   // TS + 2*2
#define SM_S 40
#define MG_T 34   // TS + 2*1
#define MG_S 36
// hysteresis tile: 5x5 window -> halo 2; origin gx = tx0-4 (4B aligned chunks)
#define HS_T 36
#define HS_S 40   // bytes per LDS row (multiple of 4)

// Flat pointers into LDS carry the LDS byte offset in addr[31:0]
// (aperture bits live in addr[63:32]); truncation yields the LDS address
// the async-DMA instructions want in their VDST operand.
__device__ __forceinline__ uint32_t lds_addr32(const void* p) {
    return (uint32_t)(uintptr_t)p;
}
__device__ __forceinline__ void async_g2l_b128(void* lds, const void* gaddr) {
    asm volatile("global_load_async_to_lds_b128 %0, %1, off"
                 :: "v"(lds_addr32(lds)), "v"(gaddr) : "memory");
}
__device__ __forceinline__ void async_g2l_b32(void* lds, const void* gaddr) {
    asm volatile("global_load_async_to_lds_b32 %0, %1, off"
                 :: "v"(lds_addr32(lds)), "v"(gaddr) : "memory");
}
__device__ __forceinline__ void wait_async0() {
    asm volatile("s_wait_asynccnt 0" ::: "memory");
}

// ---------------------------------------------------------------------------
// Kernel 1: fused gauss -> sobel -> NMS -> double threshold -> packed u8
//           bit0 = strong, bit1 = weak
// ---------------------------------------------------------------------------
__global__ __launch_bounds__(256)
void canny_fused(const float* __restrict__ x, const float* __restrict__ gk,
                 const float* __restrict__ sk, uint8_t* __restrict__ packed_out)
{
    __shared__ __align__(16) float s_in [IN_T * IN_S];
    __shared__ float s_sm [SM_T * SM_S];
    __shared__ float s_mag[MG_T * MG_S];
    __shared__ int   s_bin[MG_T * MG_S];

    const int tx0 = blockIdx.x * TS;
    const int ty0 = blockIdx.y * TS;
    const int bat = blockIdx.z;
    const int tid = threadIdx.x;
    const float* img = x + (size_t)bat * IMH * IMW;

    // --- stage 38 rows x 40 floats (origin gx = tx0-4) via async b128 DMA ---
    // chunk c of row iy covers gx in [tx0-4+4c, tx0+4c); with tx0 % 32 == 0 and
    // W = 1024, every chunk is fully in-image or fully out (no partials).
    for (int i = tid; i < IN_T * (IN_W / 4); i += 256) {
        const int iy = i / (IN_W / 4), c = i % (IN_W / 4);
        const int gy = ty0 - 3 + iy;
        const int gx = tx0 - 4 + 4 * c;
        float* dst = &s_in[iy * IN_S + 4 * c];
        if (gy >= 0 && gy < IMH && gx >= 0 && gx + 3 < IMW)
            async_g2l_b128(dst, img + (size_t)gy * IMW + gx);
        else
            *(float4*)dst = make_float4(0.f, 0.f, 0.f, 0.f);  // 'SAME' zero pad
    }
    // conv weights (uniform, tiny)
    float kg[9], ksx[9], ksy[9];
#pragma unroll
    for (int i = 0; i < 9; ++i) kg[i] = gk[i];
#pragma unroll
    for (int i = 0; i < 9; ++i) { ksx[i] = sk[2 * i]; ksy[i] = sk[2 * i + 1]; }

    wait_async0();
    __syncthreads();

    // --- 3x3 Gaussian into s_sm (halo 2); s_in x index = (tile x) + 1 ---
    for (int i = tid; i < SM_T * SM_T; i += 256) {
        const int sy = i / SM_T, sx = i % SM_T;
        float acc = 0.0f;
#pragma unroll
        for (int ky = 0; ky < 3; ++ky)
#pragma unroll
            for (int kx = 0; kx < 3; ++kx)
                acc += kg[ky * 3 + kx] * s_in[(sy + ky) * IN_S + (sx + kx) + 1];
        s_sm[sy * SM_S + sx] = acc;
    }
    __syncthreads();

    // --- 3x3 Sobel -> magnitude + direction bin (halo 1) ---
    for (int i = tid; i < MG_T * MG_T; i += 256) {
        const int my = i / MG_T, mx = i % MG_T;
        const int gy = ty0 - 1 + my, gx = tx0 - 1 + mx;
        float gxv = 0.0f, gyv = 0.0f;
#pragma unroll
        for (int ky = 0; ky < 3; ++ky)
#pragma unroll
            for (int kx = 0; kx < 3; ++kx) {
                const float v = s_sm[(my + ky) * SM_S + (mx + kx)];
                gxv += ksx[ky * 3 + kx] * v;
                gyv += ksy[ky * 3 + kx] * v;
            }
        float mag = sqrtf(gxv * gxv + gyv * gyv);
        mag = fminf(fmaxf(mag, 0.0f), 255.0f);
        float th = atan2f(gyv, gxv) * 57.29577951308232f + 90.0f; // degrees
        if (th < 0.0f)    th += 180.0f;    // theta mod 180
        if (th >= 180.0f) th -= 180.0f;
        int bn;
        if (th >= 157.5f || th <= 22.5f) bn = 0;
        else if (th < 67.5f)             bn = 1;
        else if (th < 112.5f)            bn = 2;
        else                             bn = 3;
        const bool inimg = (gy >= 0 && gy < IMH && gx >= 0 && gx < IMW);
        s_mag[my * MG_S + mx] = mag;
        s_bin[my * MG_S + mx] = inimg ? bn : -1;  // -1 => -inf pad in dilation
    }
    __syncthreads();

    // --- directional NMS + double threshold, packed output ---
    const int oy0[4] = { 0, -1, -1, -1 }, oy1[4] = { 0, 1, 1, 1 };
    const int ox0[4] = { -1, 1,  0, -1 }, ox1[4] = { 1, -1, 0, 1 };
    for (int i = tid; i < TS * TS; i += 256) {
        const int py = i / TS, px = i % TS;
        const int my = py + 1, mx = px + 1;
        const float mag0 = s_mag[my * MG_S + mx];
        const int   bin0 = s_bin[my * MG_S + mx];   // center always in-image
        float ec = 0.0f;
#pragma unroll
        for (int c = 0; c < 4; ++c) {
            const float r0 = (bin0 == c) ? mag0 : 0.0f;
            const int n1 = (my + oy0[c]) * MG_S + (mx + ox0[c]);
            const int n2 = (my + oy1[c]) * MG_S + (mx + ox1[c]);
            const int b1 = s_bin[n1], b2 = s_bin[n2];
            const float r1 = (b1 < 0) ? -__builtin_inff() : ((b1 == c) ? s_mag[n1] : 0.0f);
            const float r2 = (b2 < 0) ? -__builtin_inff() : ((b2 == c) ? s_mag[n2] : 0.0f);
            const float dil = fmaxf(r0, fmaxf(r1, r2));
            const float sup = (dil == r0) ? mag0 : 0.0f;  // exact-equality NMS
            ec = fmaxf(ec, sup);
        }
        const size_t gidx = (size_t)bat * IMH * IMW + (size_t)(ty0 + py) * IMW + (tx0 + px);
        const int strong = (ec >= 80.0f) ? 1 : 0;
        const int weak   = (ec >= 50.0f && ec < 80.0f) ? 1 : 0;
        packed_out[gidx] = (uint8_t)(strong | (weak << 1));
    }
}

// ---------------------------------------------------------------------------
// Kernel 2: hysteresis iteration on packed bytes:
//   strong' = strong | (weak & max5x5(strong));  weak bit preserved.
// ---------------------------------------------------------------------------
__global__ __launch_bounds__(256)
void canny_hyst(const uint8_t* __restrict__ p_in, uint8_t* __restrict__ p_out,
                float* __restrict__ f_out, int last)
{
    __shared__ __align__(16) uint8_t s_s[HS_T * HS_S];
    const int tx0 = blockIdx.x * TS;
    const int ty0 = blockIdx.y * TS;
    const int bat = blockIdx.z;
    const int tid = threadIdx.x;
    const uint8_t* sp = p_in + (size_t)bat * IMH * IMW;

    // stage 36 rows x 40 bytes (origin gx = tx0-4) via async b32 DMA; chunks
    // are 4B-aligned and fully in/out of the image (tx0 % 32 == 0, W = 1024).
    for (int i = tid; i < HS_T * (HS_S / 4); i += 256) {
        const int iy = i / (HS_S / 4), c = i % (HS_S / 4);
        const int gy = ty0 - 2 + iy;
        const int gx = tx0 - 4 + 4 * c;
        uint8_t* dst = &s_s[iy * HS_S + 4 * c];
        if (gy >= 0 && gy < IMH && gx >= 0 && gx + 3 < IMW)
            async_g2l_b32(dst, sp + (size_t)gy * IMW + gx);
        else
            *(uint32_t*)dst = 0u;  // clipped window: 0 never raises the max
    }
    wait_async0();
    __syncthreads();

    for (int i = tid; i < TS * TS; i += 256) {
        const int py = i / TS, px = i % TS;
        const int cy = py + 2, cx = px + 4;   // LDS x origin is gx = tx0-4
        int m = 0;
#pragma unroll
        for (int dy = -2; dy <= 2; ++dy)
#pragma unroll
            for (int dx = -2; dx <= 2; ++dx)
                m |= s_s[(cy + dy) * HS_S + (cx + dx)];
        m &= 1;                                // pooled strong (binary max)
        const int sc = s_s[cy * HS_S + cx];
        const int ns = (sc & 1) | ((sc >> 1) & m);
        const size_t gidx = (size_t)bat * IMH * IMW + (size_t)(ty0 + py) * IMW + (tx0 + px);
        if (last) f_out[gidx] = (float)ns;
        else      p_out[gidx] = (uint8_t)(ns | (sc & 2));  // keep weak bit
    }
}

extern "C" void kernel_launch(void* const* d_in, const int* in_sizes, int n_in,
                              void* d_out, int out_size, void* d_ws, size_t ws_size,
                              hipStream_t stream) {
    const float* x  = (const float*)d_in[0];
    const float* gk = (const float*)d_in[1];
    const float* sk = (const float*)d_in[2];
    float* out = (float*)d_out;

    const int B = in_sizes[0] / (IMH * IMW);
    const size_t N = (size_t)B * IMH * IMW;

    uint8_t* pA = (uint8_t*)d_ws;      // packed strong/weak ping
    uint8_t* pB = pA + N;              // packed strong/weak pong

    dim3 grid(IMW / TS, IMH / TS, B);
    dim3 block(256, 1, 1);

    canny_fused<<<grid, block, 0, stream>>>(x, gk, sk, pA);
    canny_hyst <<<grid, block, 0, stream>>>(pA, pB, nullptr, 0);
    canny_hyst <<<grid, block, 0, stream>>>(pB, pA, nullptr, 0);
    canny_hyst <<<grid, block, 0, stream>>>(pA, nullptr, out, 1);
}